// HierDDpm_68719477575
// MI455X (gfx1250) — compile-verified
//
#include <hip/hip_runtime.h>

#define DD 256
#define BN 8
#define SN 1024
#define EPSF 1e-5f
#define TWO_PI 6.28318530717958647692f

typedef float v2f __attribute__((ext_vector_type(2)));
typedef float v8f __attribute__((ext_vector_type(8)));

// ---------------------------------------------------------------------------
// Kernel 0: Rtab[i*D+j] = 2*pi / (i*D + j + 2)   (64K entries, 256 KB)
//   Position- and layer-independent; computed once, stays hot in L2.
// ---------------------------------------------------------------------------
__global__ __launch_bounds__(256) void period_table_kernel(float* __restrict__ Rtab) {
  int idx = blockIdx.x * 256 + threadIdx.x;  // 0..65535
  Rtab[idx] = TWO_PI / (float)(idx + 2);     // full-precision divide, one-time
}

// ---------------------------------------------------------------------------
// Kernel 1: T = LayerNorm(X @ M^T)
//   X: [B*S, D] row-major, M: [D, D] row-major, T: [B*S, D]
//   Block = 512 threads (16 waves). Block handles 16 rows of X.
//   Wave w computes the 16-col output tile w via V_WMMA_F32_16X16X4_F32,
//   then wave w LayerNorms row w.
// ---------------------------------------------------------------------------
__global__ __launch_bounds__(512) void gemm_ln_kernel(
    const float* __restrict__ X, const float* __restrict__ M,
    const float* __restrict__ g, const float* __restrict__ bia,
    float* __restrict__ T) {
  __shared__ float Xs[16][DD];
  __shared__ float Ts[16][DD];

  const int tid  = threadIdx.x;      // 0..511
  const int wave = tid >> 5;         // 0..15
  const int lane = tid & 31;
  const int row0 = blockIdx.x * 16;  // base row of this block

  // Cooperative, coalesced load of 16x256 X tile (8 floats per thread).
  #pragma unroll
  for (int e = 0; e < 8; ++e) {
    int idx = tid + 512 * e;   // 0..4095
    int r = idx >> 8;
    int c = idx & 255;
    Xs[r][c] = X[(size_t)(row0 + r) * DD + c];
  }
  __syncthreads();

  // WMMA f32 16x16x4 fragment indices (wave32):
  //   A (16x4): lane m = L&15 holds rows; K = (L>>4)*2 + vgpr
  //   B (4x16): lane n = L&15 holds cols; K = (L>>4)*2 + vgpr
  const int i0 = wave * 16;          // output col tile
  const int mn = lane & 15;
  const int kh = (lane >> 4) * 2;

  v8f acc = {};
  for (int k0 = 0; k0 < DD; k0 += 4) {
    v2f a, bm;
    a.x = Xs[mn][k0 + kh];
    a.y = Xs[mn][k0 + kh + 1];
    // B[k][n] = M[i0+n][k0+k]  (GEMM is X @ M^T)
    const float* mp = M + (size_t)(i0 + mn) * DD + (k0 + kh);
    bm.x = mp[0];
    bm.y = mp[1];
    acc = __builtin_amdgcn_wmma_f32_16x16x4_f32(false, a, false, bm,
                                                (short)0, acc, false, false);
  }

  // D layout: VGPR r, lanes 0-15 -> row r, lanes 16-31 -> row r+8; col = lane&15
  {
    int mrow = (lane < 16) ? 0 : 8;
    #pragma unroll
    for (int r = 0; r < 8; ++r)
      Ts[mrow + r][i0 + mn] = acc[r];
  }
  __syncthreads();

  // LayerNorm: wave w normalizes row w (population variance, as jnp.var).
  float s1 = 0.f, s2 = 0.f;
  #pragma unroll
  for (int e = 0; e < 8; ++e) {
    float v = Ts[wave][lane + 32 * e];
    s1 += v;
    s2 += v * v;
  }
  #pragma unroll
  for (int off = 16; off > 0; off >>= 1) {
    s1 += __shfl_xor(s1, off, 32);
    s2 += __shfl_xor(s2, off, 32);
  }
  const float mu  = s1 * (1.0f / DD);
  const float var = s2 * (1.0f / DD) - mu * mu;
  const float rs  = rsqrtf(var + EPSF);
  #pragma unroll
  for (int e = 0; e < 8; ++e) {
    int c = lane + 32 * e;
    T[(size_t)(row0 + wave) * DD + c] = (Ts[wave][c] - mu) * rs * g[c] + bia[c];
  }
}

// ---------------------------------------------------------------------------
// Kernel 2: Y[b,s,i] = sum_j T[b,s,j] * P[i,j] * cos(s * Rtab[i*D+j]) + X[b,s,i]
//   One block per position s (1024 blocks), 256 threads = 8 waves.
//   Wave w owns i-tiles {w, w+8}. B fragments are synthesized on the fly
//   (one v_cos_f32 per weight element, reciprocal period from L2 table)
//   and fed to f32 WMMA; batch rows padded 8->16 so EXEC stays all-1s.
// ---------------------------------------------------------------------------
__global__ __launch_bounds__(256) void phase_gemm_kernel(
    const float* __restrict__ T, const float* __restrict__ P,
    const float* __restrict__ Rtab,
    const float* __restrict__ X, float* __restrict__ Y) {
  __shared__ float Ts[16][DD];  // rows 8..15 stay zero (A-tile padding)

  const int s    = blockIdx.x;
  const int tid  = threadIdx.x;  // 0..255
  const int wave = tid >> 5;     // 0..7
  const int lane = tid & 31;

  // Load 8 T-rows for this position; zero the padding rows.
  #pragma unroll
  for (int e = 0; e < 16; ++e) {
    int idx = tid + 256 * e;  // 0..4095
    int r = idx >> 8;
    int c = idx & 255;
    Ts[r][c] = (r < BN) ? T[((size_t)r * SN + s) * DD + c] : 0.0f;
  }
  __syncthreads();

  const float kf = (float)s;
  const int mn = lane & 15;
  const int kh = (lane >> 4) * 2;

  for (int t = 0; t < 2; ++t) {
    const int i0   = (wave + t * 8) * 16;
    const int irow = i0 + mn;  // global output index i for this lane's B col
    v8f acc = {};
    for (int k0 = 0; k0 < DD; k0 += 4) {
      const int j   = k0 + kh;
      const int idx = irow * DD + j;
      v2f a, bm;
      a.x = Ts[mn][j];
      a.y = Ts[mn][j + 1];
      // B[k][n] = P[i][j] * cos(2*pi*s / (i*D + j + 2)); 2pi/p from table
      const float* pp = P + idx;
      const float* rr = Rtab + idx;
      bm.x = pp[0] * __cosf(kf * rr[0]);
      bm.y = pp[1] * __cosf(kf * rr[1]);
      acc = __builtin_amdgcn_wmma_f32_16x16x4_f32(false, a, false, bm,
                                                  (short)0, acc, false, false);
    }
    // Rows 0..7 of D live in acc[0..7] on lanes 0..15. Add residual, store.
    if (lane < 16) {
      #pragma unroll
      for (int r = 0; r < BN; ++r) {
        size_t o = ((size_t)r * SN + s) * DD + i0 + lane;
        Y[o] = acc[r] + X[o];
      }
    }
  }
}

// ---------------------------------------------------------------------------
// Host launcher. Inputs: seq, M1, P1, g1, b1, M2, P2, g2, b2 (all f32).
// Workspace: T buffer [B*S, D] f32 (8 MB) + Rtab [D*D] f32 (256 KB).
// ---------------------------------------------------------------------------
extern "C" void kernel_launch(void* const* d_in, const int* in_sizes, int n_in,
                              void* d_out, int out_size, void* d_ws, size_t ws_size,
                              hipStream_t stream) {
  (void)in_sizes; (void)n_in; (void)out_size; (void)ws_size;
  const float* seq = (const float*)d_in[0];
  const float* M1  = (const float*)d_in[1];
  const float* P1  = (const float*)d_in[2];
  const float* g1  = (const float*)d_in[3];
  const float* b1  = (const float*)d_in[4];
  const float* M2  = (const float*)d_in[5];
  const float* P2  = (const float*)d_in[6];
  const float* g2  = (const float*)d_in[7];
  const float* b2  = (const float*)d_in[8];
  float* out  = (float*)d_out;
  float* Tws  = (float*)d_ws;                          // [B*S, D] f32, 8 MB
  float* Rtab = (float*)((char*)d_ws + (size_t)BN * SN * DD * sizeof(float));

  const int rowBlocks = (BN * SN) / 16;  // 512

  // One-time reciprocal-period table (deterministic every call).
  period_table_kernel<<<(DD * DD) / 256, 256, 0, stream>>>(Rtab);

  // Layer 1: T = LN(seq @ M1^T); out = phase(T, P1) + seq
  gemm_ln_kernel<<<rowBlocks, 512, 0, stream>>>(seq, M1, g1, b1, Tws);
  phase_gemm_kernel<<<SN, 256, 0, stream>>>(Tws, P1, Rtab, seq, out);

  // Layer 2: T = LN(out @ M2^T); out = phase(T, P2) + out  (in-place residual)
  gemm_ln_kernel<<<rowBlocks, 512, 0, stream>>>(out, M2, g2, b2, Tws);
  phase_gemm_kernel<<<SN, 256, 0, stream>>>(Tws, P2, Rtab, out, out);
}